// NearestNeighborSparseLayer_53953379172899
// MI455X (gfx1250) — compile-verified
//
#include <hip/hip_runtime.h>
#include <hip/hip_bf16.h>

#define IN_F   4096
#define OUT_F  4096
#define TOKENS 8192

typedef __attribute__((ext_vector_type(2))) float v2f;
typedef __attribute__((ext_vector_type(8))) float v8f;

// ---------------------------------------------------------------------------
// Kernel 1: collapse connections * tridiagonal_mask * weight.T into the three
// nonzero diagonals of eff [in, out]:
//   lo[j]  = eff[j-1, j],  mid[j] = eff[j, j],  hi[j] = eff[j+1, j]
// conn is [IN, OUT] row-major, w is [OUT, IN] row-major, nn is [IN, IN].
// ---------------------------------------------------------------------------
__global__ void build_diags_kernel(const float* __restrict__ conn,
                                   const float* __restrict__ nn,
                                   const float* __restrict__ w,
                                   float* __restrict__ lo,
                                   float* __restrict__ mid,
                                   float* __restrict__ hi) {
    int j = blockIdx.x * blockDim.x + threadIdx.x;
    if (j >= OUT_F) return;
    float l = 0.f, h = 0.f;
    float m = conn[(size_t)j * OUT_F + j] * nn[(size_t)j * IN_F + j]
            * w[(size_t)j * IN_F + j];
    if (j >= 1)
        l = conn[(size_t)(j - 1) * OUT_F + j] * nn[(size_t)(j - 1) * IN_F + j]
          * w[(size_t)j * IN_F + (j - 1)];
    if (j < IN_F - 1)
        h = conn[(size_t)(j + 1) * OUT_F + j] * nn[(size_t)(j + 1) * IN_F + j]
          * w[(size_t)j * IN_F + (j + 1)];
    lo[j] = l;  mid[j] = m;  hi[j] = h;
}

// ---------------------------------------------------------------------------
// Kernel 2: banded mini-GEMM stencil with V_WMMA_F32_16X16X4_F32.
// Each wave owns one 16-column feature tile (j0) and loops over token tiles.
// Output tile needs input rows j0-2 .. j0+17 -> five K=4 blocks -> 5 WMMAs.
// ---------------------------------------------------------------------------
__global__ void __launch_bounds__(256)
stencil_wmma_kernel(const float* __restrict__ x,
                    const float* __restrict__ lo,
                    const float* __restrict__ mid,
                    const float* __restrict__ hi,
                    const float* __restrict__ bias,
                    float* __restrict__ out) {
    const int lane = threadIdx.x & 31;
    const int wave = threadIdx.x >> 5;
    const int j0   = blockIdx.y << 4;          // feature tile base column
    const int col  = j0 + (lane & 15);         // this lane's N column
    const int koff = (lane >> 4) << 1;         // 0 (lanes 0-15) / 2 (lanes 16-31)

    // --- Build banded B fragments once per feature tile (reused over tokens).
    // B block t holds eff[kb + koff + s, col], kb = j0 - 2 + 4t.
    const float cl = lo[col], cm = mid[col], ch = hi[col];
    v2f Bfrag[5];
#pragma unroll
    for (int t = 0; t < 5; ++t) {
        const int ib = j0 - 2 + 4 * t + koff;
#pragma unroll
        for (int s = 0; s < 2; ++s) {
            const int i = ib + s;
            float b = 0.f;
            if (i == col - 1)      b = cl;   // lo[0]  == 0 handles i = -1
            else if (i == col)     b = cm;
            else if (i == col + 1) b = ch;   // hi[N-1] == 0 handles i = N
            Bfrag[t][s] = b;
        }
    }

    const float bv = bias[col];                // C layout: N = lane&15 for all 8 regs

    const int gw     = blockIdx.x * 8 + wave;  // global wave id within this column
    const int stride = gridDim.x * 8;

    for (int tt = gw; tt < TOKENS / 16; tt += stride) {
        const int m0  = tt << 4;
        const int row = m0 + (lane & 15);      // A layout: M = lane&15
        const float* xr = x + (size_t)row * IN_F;

        v8f c;
#pragma unroll
        for (int v = 0; v < 8; ++v) c[v] = bv; // fold bias into accumulator

#pragma unroll
        for (int t = 0; t < 5; ++t) {
            int i  = j0 - 2 + 4 * t + koff;    // even -> 8B-aligned b64 load
            int ic = i < 0 ? 0 : (i > IN_F - 2 ? IN_F - 2 : i); // clamp; B==0 there
            v2f a  = *(const v2f*)(xr + ic);   // {x[row,ic], x[row,ic+1]}
            c = __builtin_amdgcn_wmma_f32_16x16x4_f32(
                    false, a, false, Bfrag[t], (short)0, c, false, false);
        }

        // D layout: reg v, lane l -> out[m0 + v + 8*(l>=16), j0 + (l&15)]
        float* orow = out + (size_t)(m0 + ((lane >> 4) << 3)) * OUT_F + col;
#pragma unroll
        for (int v = 0; v < 8; ++v)
            __builtin_nontemporal_store(c[v], orow + (size_t)v * OUT_F);
    }
}

// ---------------------------------------------------------------------------
extern "C" void kernel_launch(void* const* d_in, const int* in_sizes, int n_in,
                              void* d_out, int out_size, void* d_ws, size_t ws_size,
                              hipStream_t stream) {
    const float* x    = (const float*)d_in[0];  // [8192, 4096]
    const float* conn = (const float*)d_in[1];  // [4096, 4096] (in, out)
    const float* nn   = (const float*)d_in[2];  // [4096, 4096] tridiagonal mask
    const float* w    = (const float*)d_in[3];  // [4096, 4096] (out, in)
    const float* bias = (const float*)d_in[4];  // [4096]
    float* out = (float*)d_out;

    float* lo  = (float*)d_ws;                  // 3 * 4096 floats = 48 KB scratch
    float* mid = lo + OUT_F;
    float* hi  = mid + OUT_F;

    build_diags_kernel<<<(OUT_F + 255) / 256, 256, 0, stream>>>(conn, nn, w, lo, mid, hi);

    dim3 grid(8, OUT_F / 16);                   // 8 blocks * 8 waves = 64 waves/column,
    stencil_wmma_kernel<<<grid, 256, 0, stream>>>(x, lo, mid, hi, bias, out);
}